// PairwiseLoss_62955630624882
// MI455X (gfx1250) — compile-verified
//
#include <hip/hip_runtime.h>
#include <stdint.h>

#define L 16
#define TI 32                    // tile rows for both i and j
#define LDS_STRIDE 17            // padded LDS row stride (floats): conflict-free columns
#define TILE_FLOATS (TI * LDS_STRIDE)   // 544
#define EPSV 1e-10f

typedef __attribute__((ext_vector_type(2)))  float        v2f;
typedef __attribute__((ext_vector_type(8)))  float        v8f;
typedef __attribute__((ext_vector_type(4)))  unsigned int v4u;
typedef __attribute__((ext_vector_type(8)))  int          v8i;
typedef __attribute__((ext_vector_type(4)))  int          v4i;

// TDM: DMA a TI x L f32 tile (contiguous rows of a row-major [N][L] matrix) from
// global memory into LDS. D# uses pad_enable with pad_interval=16 DWORDs /
// pad_amount=1 DWORD so rows land in LDS at stride 17 floats (bank-conflict-free
// for per-lane row reads with lane-stride 17).
__device__ __forceinline__ void tdm_load_tile(const float* gsrc,
                                              uint32_t lds_byte_off,
                                              uint32_t nrows_total) {
  uint64_t ga = (uint64_t)(uintptr_t)gsrc;
  v4u g0;
  g0.x = 1u;                                                  // count=1, user descriptor
  g0.y = lds_byte_off;                                        // D#.lds_addr (bytes)
  g0.z = (uint32_t)ga;                                        // global_addr[31:0]
  g0.w = (uint32_t)((ga >> 32) & 0x01FFFFFFu) | (2u << 30);   // global_addr[56:32] | type=2

  uint32_t dim0   = L;             // tensor fastest dim (elements)
  uint32_t dim1   = nrows_total;   // tensor rows
  uint32_t tile0  = L;             // tile width  (elements)
  uint32_t tile1  = TI;            // tile height (rows)
  uint64_t strd0  = L;             // row stride (elements)
  uint32_t w0 = (2u << 16)         // data_size = 4 bytes
              | (1u << 20)         // pad_enable (LDS-dest padding on load)
              | (3u << 22);        // pad_interval code 3 = every 16 DWORDs
                                   // pad_amount  code 0 = 1 DWORD (bits 31:25 zero)
  v8i g1;
  g1[0] = (int)w0;
  g1[1] = (int)((dim0 & 0xFFFFu) << 16);                            // tensor_dim0[15:0]
  g1[2] = (int)(((dim0 >> 16) & 0xFFFFu) | ((dim1 & 0xFFFFu) << 16));
  g1[3] = (int)(((dim1 >> 16) & 0xFFFFu) | ((tile0 & 0xFFFFu) << 16));
  g1[4] = (int)(tile1 & 0xFFFFu);                                   // tile_dim1 | tile_dim2=0
  g1[5] = (int)(uint32_t)strd0;                                     // tensor_dim0_stride[31:0]
  g1[6] = (int)(uint32_t)((strd0 >> 32) & 0xFFFFu);                 // stride hi | dim1_stride=0
  g1[7] = 0;

  v4i gz = (v4i)0;
#if __clang_major__ >= 23
  v8i gz8 = (v8i)0;
  __builtin_amdgcn_tensor_load_to_lds(g0, g1, gz, gz, gz8, 0);
#else
  __builtin_amdgcn_tensor_load_to_lds(g0, g1, gz, gz, 0);
#endif
}

// softplus(o) = ln(1+e^o) = ln2 * log2(1 + 2^(o*log2e)); o pre-clipped to +-50
__device__ __forceinline__ float softplus_fast(float o) {
  float e = exp2f(o * 1.44269504089f);     // v_exp_f32
  return 0.69314718056f * __log2f(1.0f + e);  // v_log_f32
}

__global__ __launch_bounds__(256) void ranknet_tiles(
    const float* __restrict__ input, const float* __restrict__ target,
    const float* __restrict__ lwg, float* __restrict__ partials,
    int T, int N) {
  __shared__ __align__(16) float lds[4 * TILE_FLOATS + L + 256];
  float* xi  = lds;                       // input rows, i-tile (stride 17)
  float* xj  = lds + 1 * TILE_FLOATS;     // input rows, j-tile
  float* ui  = lds + 2 * TILE_FLOATS;     // target rows, i-tile
  float* uj  = lds + 3 * TILE_FLOATS;     // target rows, j-tile
  float* lw  = lds + 4 * TILE_FLOATS;     // label weights
  float* red = lw + L;                    // 256-float reduction buffer

  // decode upper-triangular tile pair (ti <= tj) from linear block id (uniform)
  int p = blockIdx.x, ti = 0;
  while (p >= T - ti) { p -= T - ti; ++ti; }
  int tj = ti + p;
  const int i0 = ti * TI, j0 = tj * TI;

  const int tid = threadIdx.x;
  if (tid < 32) {  // wave 0 issues the 4 tensor DMAs (per-wave op, EXEC ignored)
    uint32_t base = (uint32_t)(uintptr_t)lds;  // low 32 bits of flat addr = LDS offset
    tdm_load_tile(input  + (size_t)i0 * L, base + 0u * TILE_FLOATS * 4u, (uint32_t)N);
    tdm_load_tile(input  + (size_t)j0 * L, base + 1u * TILE_FLOATS * 4u, (uint32_t)N);
    tdm_load_tile(target + (size_t)i0 * L, base + 2u * TILE_FLOATS * 4u, (uint32_t)N);
    tdm_load_tile(target + (size_t)j0 * L, base + 3u * TILE_FLOATS * 4u, (uint32_t)N);
    __builtin_amdgcn_s_wait_tensorcnt(0);
  }
  if (tid < L) lw[tid] = lwg[tid];
  __syncthreads();

  const bool diag = (ti == tj);
  float acc = 0.0f;
  #pragma unroll
  for (int k = 0; k < (TI * TI) / 256; ++k) {
    int q  = tid + k * 256;
    int li = q >> 5;          // wave-uniform -> xi/ui reads are broadcasts
    int lj = q & 31;          // lane id -> xj/uj reads stride-17, conflict-free
    if (!diag || lj > li) {
      const float* xr = xi + li * LDS_STRIDE;
      const float* xc = xj + lj * LDS_STRIDE;
      const float* tr = ui + li * LDS_STRIDE;
      const float* tc = uj + lj * LDS_STRIDE;
      #pragma unroll
      for (int l = 0; l < L; ++l) {
        float o = xr[l] - xc[l];
        o = fminf(fmaxf(o, -50.0f), 50.0f);
        float d = tr[l] - tc[l];                    // in {-1, 0, +1}
        float sp = softplus_fast(o);
        float pt = (d > 0.0f) ? (1.0f - EPSV) : EPSV;  // clip(0.5*(1+d), eps, 1-eps)
        float term = (sp - pt * o) * lw[l];
        acc += (d != 0.0f) ? term : 0.0f;           // mask: differing labels only
      }
    }
  }

  // block reduction: 256 -> 32 via LDS, then 32 -> 1 via exact f32 WMMA
  red[tid] = acc;
  __syncthreads();
  if (tid < 32) {  // wave 0, EXEC all-ones as WMMA requires
    float s = 0.0f;
    #pragma unroll
    for (int r = 0; r < 8; ++r) s += red[tid + r * 32];
    // A[16x4]: VGPR0 = lane partial (K0 for lanes 0-15, K2 for 16-31), VGPR1 = 0
    v2f a; a.x = s;    a.y = 0.0f;
    // B[4x16] = all ones -> D[m][n] = s[m] + s[m+16], replicated over n (exact f32)
    v2f b; b.x = 1.0f; b.y = 1.0f;
    v8f c = {};
    v8f dm = __builtin_amdgcn_wmma_f32_16x16x4_f32(false, a, false, b,
                                                   (short)0, c, false, false);
    // lanes 0-15 hold rows 0..7 in dm[0..7]; lanes 16-31 hold rows 8..15
    float t = dm[0] + dm[1] + dm[2] + dm[3] + dm[4] + dm[5] + dm[6] + dm[7];
    t += __shfl_xor(t, 16, 32);   // combine the two row halves
    if (tid == 0) partials[blockIdx.x] = t;
  }
}

__global__ __launch_bounds__(256) void final_reduce(
    const float* __restrict__ partials, int n, float* __restrict__ out) {
  __shared__ float red[256];
  float s = 0.0f;
  for (int i = threadIdx.x; i < n; i += 256) s += partials[i];
  red[threadIdx.x] = s;
  __syncthreads();
  for (int off = 128; off > 0; off >>= 1) {
    if (threadIdx.x < off) red[threadIdx.x] += red[threadIdx.x + off];
    __syncthreads();
  }
  if (threadIdx.x == 0) out[0] = red[0] / (float)L;
}

extern "C" void kernel_launch(void* const* d_in, const int* in_sizes, int n_in,
                              void* d_out, int out_size, void* d_ws, size_t ws_size,
                              hipStream_t stream) {
  const float* input  = (const float*)d_in[0];   // [N, L] f32
  const float* target = (const float*)d_in[1];   // [N, L] f32 (binary)
  const float* lw     = (const float*)d_in[2];   // [L] f32
  float* out = (float*)d_out;                    // scalar f32 loss

  int N = in_sizes[0] / L;        // 2048
  int T = N / TI;                 // 64 tiles
  int nblocks = T * (T + 1) / 2;  // 2080 upper-triangular tile pairs

  float* partials = (float*)d_ws; // nblocks floats of scratch

  ranknet_tiles<<<nblocks, 256, 0, stream>>>(input, target, lw, partials, T, N);
  final_reduce<<<1, 256, 0, stream>>>(partials, nblocks, out);
}